// DPL_synthetic_46394236731763
// MI455X (gfx1250) — compile-verified
//
#include <hip/hip_runtime.h>
#include <math.h>

typedef __attribute__((ext_vector_type(2))) float v2f;
typedef __attribute__((ext_vector_type(8))) float v8f;

#define HID   128
#define NFACT 5
#define NCLS  13
#define SW2S  (HID + 2)   // padded row stride: even (keeps 8B align), rotates LDS banks

// One wave (32 lanes) fully processes 16 batch elements.
// WMMA orientation: D(16x16) = A(16x4: W2^T tile, M=class) x B(4x16: H^T tile, N=batch)
// => lane l (l<16) holds cs[class r][batch l] in acc[r], r=0..4 -> lane-local softmax.
__global__ __launch_bounds__(256) void dpl_wmma_kernel(
    const float* __restrict__ z,    // [B,3,1]
    const float* __restrict__ W1,   // [1,128]
    const float* __restrict__ b1,   // [128]
    const float* __restrict__ W2,   // [128,5]
    const float* __restrict__ b2,   // [5]
    float* __restrict__ out_cs,     // [B,3,5]
    float* __restrict__ out_py,     // [B,13]
    float* __restrict__ out_pcs,    // [B,3,5]
    int B)
{
    // W2^T zero-padded to 16 rows: unconditional A-fragment loads (no EXEC dance).
    __shared__ float sW2T[16 * SW2S];
    // Interleaved {W1[k],W1[k+1],b1[k],b1[k+1]} quads: one b128 load per K-pair.
    __shared__ float sWB[(HID / 2) * 4];
    __shared__ float sB2[NFACT];

    for (int i = threadIdx.x; i < 16 * SW2S; i += blockDim.x) {
        const int mm = i / SW2S, kk = i % SW2S;
        sW2T[i] = (mm < NFACT && kk < HID) ? W2[kk * NFACT + mm] : 0.f;
    }
    for (int i = threadIdx.x; i < HID / 2; i += blockDim.x) {
        sWB[i * 4 + 0] = W1[i * 2 + 0];
        sWB[i * 4 + 1] = W1[i * 2 + 1];
        sWB[i * 4 + 2] = b1[i * 2 + 0];
        sWB[i * 4 + 3] = b1[i * 2 + 1];
    }
    if (threadIdx.x < NFACT) sB2[threadIdx.x] = b2[threadIdx.x];
    __syncthreads();

    const int lane  = threadIdx.x & 31;
    const int wave  = threadIdx.x >> 5;
    const int col   = lane & 15;   // batch column (B/D) == class row (A)
    const int khalf = lane >> 4;   // K sub-offset: lanes 0-15 -> K{0,1}, 16-31 -> K{2,3}

    const int b  = blockIdx.x * 128 + wave * 16 + col;
    const int bc = (b < B) ? b : (B - 1);          // clamped for safe loads
    const bool valid = (b < B) && (lane < 16);

    const float* aRow = &sW2T[col * SW2S];          // this lane's A-matrix row (class col)

    float pd[3][NFACT];                             // pCs for this lane's batch element

    for (int d = 0; d < 3; ++d) {
        const float zv = z[bc * 3 + d];
        v8f acc = {0.f, 0.f, 0.f, 0.f, 0.f, 0.f, 0.f, 0.f};

        #pragma unroll 4
        for (int kt = 0; kt < HID / 4; ++kt) {
            const int k0 = kt * 4 + khalf * 2;                 // even -> aligned
            const v2f    a  = *(const v2f*)(aRow + k0);        // W2^T[col][k0..k0+1]
            const float4 wb = *(const float4*)(&sWB[k0 * 2]);  // {W1 pair, b1 pair}
            v2f bf;
            bf.x = fmaxf(fmaf(zv, wb.x, wb.z), 0.f);           // h[k0]   (batch col)
            bf.y = fmaxf(fmaf(zv, wb.y, wb.w), 0.f);           // h[k0+1]
            acc = __builtin_amdgcn_wmma_f32_16x16x4_f32(
                false, a, false, bf, (short)0, acc, false, false);
        }

        // Lane-local epilogue: classes 0..4 live in acc[0..4] for lanes 0..15.
        float c[NFACT], p[NFACT];
        float mx = -3.402823466e+38f;
        #pragma unroll
        for (int i = 0; i < NFACT; ++i) { c[i] = acc[i] + sB2[i]; mx = fmaxf(mx, c[i]); }
        float s = 0.f;
        #pragma unroll
        for (int i = 0; i < NFACT; ++i) { p[i] = __expf(c[i] - mx); s += p[i]; }
        const float inv = 1.f / s;
        float s2 = 0.f;
        #pragma unroll
        for (int i = 0; i < NFACT; ++i) { p[i] = fmaf(p[i], inv, 1e-5f); s2 += p[i]; }
        const float inv2 = 1.f / s2;
        #pragma unroll
        for (int i = 0; i < NFACT; ++i) { p[i] *= inv2; pd[d][i] = p[i]; }

        if (valid) {
            float* cso = out_cs  + (size_t)(b * 3 + d) * NFACT;
            float* pco = out_pcs + (size_t)(b * 3 + d) * NFACT;
            #pragma unroll
            for (int i = 0; i < NFACT; ++i) { cso[i] = c[i]; pco[i] = p[i]; }
        }
    }

    // worlds_prob @ w_q == convolution of the three digit distributions by sum.
    float s01[2 * NFACT - 1];
    #pragma unroll
    for (int i = 0; i < 2 * NFACT - 1; ++i) s01[i] = 0.f;
    #pragma unroll
    for (int i = 0; i < NFACT; ++i)
        #pragma unroll
        for (int j = 0; j < NFACT; ++j)
            s01[i + j] = fmaf(pd[0][i], pd[1][j], s01[i + j]);

    float qp[NCLS];
    #pragma unroll
    for (int q = 0; q < NCLS; ++q) qp[q] = 0.f;
    #pragma unroll
    for (int i = 0; i < 2 * NFACT - 1; ++i)
        #pragma unroll
        for (int k = 0; k < NFACT; ++k)
            qp[i + k] = fmaf(s01[i], pd[2][k], qp[i + k]);

    float ssum = 0.f;
    #pragma unroll
    for (int q = 0; q < NCLS; ++q) { qp[q] += 1e-5f; ssum += qp[q]; }
    const float invq = 1.f / ssum;

    if (valid) {
        float* pyo = out_py + (size_t)b * NCLS;
        #pragma unroll
        for (int q = 0; q < NCLS; ++q) pyo[q] = qp[q] * invq;
    }
}

extern "C" void kernel_launch(void* const* d_in, const int* in_sizes, int n_in,
                              void* d_out, int out_size, void* d_ws, size_t ws_size,
                              hipStream_t stream) {
    const float* z  = (const float*)d_in[0];
    const float* W1 = (const float*)d_in[1];
    const float* b1 = (const float*)d_in[2];
    const float* W2 = (const float*)d_in[3];
    const float* b2 = (const float*)d_in[4];
    // d_in[5] = w_q: deterministic one-hot digit-sum matrix, realized as convolution.

    const int B = in_sizes[0] / 3;

    float* out     = (float*)d_out;
    float* out_cs  = out;
    float* out_py  = out_cs + (size_t)B * 3 * NFACT;
    float* out_pcs = out_py + (size_t)B * NCLS;

    const int blocks = (B + 127) / 128;   // 128 batch elements per 256-thread block (8 waves x 16)
    dpl_wmma_kernel<<<blocks, 256, 0, stream>>>(z, W1, b1, W2, b2,
                                                out_cs, out_py, out_pcs, B);
}